// voxel_dgcnn_85169201480376
// MI455X (gfx1250) — compile-verified
//
#include <hip/hip_runtime.h>
#include <math.h>

typedef float v2f __attribute__((ext_vector_type(2)));
typedef float v8f __attribute__((ext_vector_type(8)));

#define NB    759
#define NPTS  220
#define KNN   10
#define EPSB  1e-5f
#define NEGINF (-3.402823466e38f)

__device__ __forceinline__ v8f wmma_f32(v2f a, v2f b, v8f c) {
  // D(16x16 f32) = A(16x4 f32) x B(4x16 f32) + C  -- exact f32 matrix op
  return __builtin_amdgcn_wmma_f32_16x16x4_f32(false, a, false, b, (short)0, c,
                                               false, false);
}

__device__ __forceinline__ float leakyf(float x) { return x >= 0.f ? x : 0.2f * x; }

// ---------------------------------------------------------------------------
// KNN: one workgroup per batch. Points live in LDS; each of 220 threads keeps
// a register top-10 of pair = 2*dot - |x_n|^2 - |x_m|^2 (strict >, so ties keep
// the earlier index, matching jax.lax.top_k stability).
// ---------------------------------------------------------------------------
template <int C>
__global__ __launch_bounds__(256) void knn_kernel(const float* __restrict__ xin,
                                                  int rs, int choff,
                                                  int* __restrict__ idx) {
  __shared__ float pts[NPTS * C];
  __shared__ float sq[NPTS];
  const int b = blockIdx.x;
  const int tid = threadIdx.x;
  const float* xb = xin + (size_t)b * NPTS * rs + choff;

  for (int i = tid; i < NPTS * C; i += 256) {
    int n = i / C, c = i - n * C;
    pts[i] = xb[n * rs + c];
  }
  __syncthreads();
  if (tid < NPTS) {
    float s = 0.f;
#pragma unroll
    for (int c = 0; c < C; ++c) { float v = pts[tid * C + c]; s += v * v; }
    sq[tid] = s;
  }
  __syncthreads();
  if (tid < NPTS) {
    float tv[KNN]; int ti[KNN];
#pragma unroll
    for (int j = 0; j < KNN; ++j) { tv[j] = NEGINF; ti[j] = 0; }
    const float sn = sq[tid];
    const int base = tid * C;
    for (int m = 0; m < NPTS; ++m) {
      float dot = 0.f;
#pragma unroll 4
      for (int c = 0; c < C; ++c) dot += pts[base + c] * pts[m * C + c];
      float d = 2.f * dot - sn - sq[m];
      if (d > tv[KNN - 1]) {
        int j = KNN - 1;
        while (j > 0 && tv[j - 1] < d) {
          tv[j] = tv[j - 1]; ti[j] = ti[j - 1]; --j;
        }
        tv[j] = d; ti[j] = m;
      }
    }
    int* op = idx + ((size_t)b * NPTS + tid) * KNN;
#pragma unroll
    for (int j = 0; j < KNN; ++j) op[j] = ti[j];
  }
}

// ---------------------------------------------------------------------------
// EdgeConv via f32 WMMA. One wave per workgroup handles 16 out-channels x
// 80 positions (8 points x 10 neighbors = 5 column tiles). Edge features are
// gathered on the fly (per-batch x fits easily in L2). After the K loop the
// tile lands in LDS and the wave does the max-over-k + leaky reduction.
// ---------------------------------------------------------------------------
template <int CIN, int COUT>
__global__ __launch_bounds__(32) void edgeconv_kernel(
    const float* __restrict__ xin, int rs_in, int choff_in,
    const float* __restrict__ W,   // (COUT, 2*CIN)
    const int* __restrict__ idx,
    float* __restrict__ xc, int choff_out) {
  constexpr int K2 = 2 * CIN;
  constexpr int KSTEPS = ((K2 + 7) / 8) * 2;  // K padded to multiple of 8
  const int NGRP = 28;                        // ceil(220/8)
  const int otile = blockIdx.x / NGRP;
  const int ngrp = blockIdx.x % NGRP;
  const int b = blockIdx.y;
  const int lane = threadIdx.x;
  const int col = lane & 15;
  const int half2 = (lane >> 4) << 1;  // K sub-offset: 0 or 2
  const int obase = otile * 16;
  const int nbase = ngrp * 8;

  __shared__ float lh[16 * 80];

  const float* xb = xin + (size_t)b * NPTS * rs_in + choff_in;
  int nb_off[5], cb_off[5];
#pragma unroll
  for (int t = 0; t < 5; ++t) {
    int pos = nbase * KNN + t * 16 + col;
    if (pos > NPTS * KNN - 1) pos = NPTS * KNN - 1;  // keep EXEC full for WMMA
    int n = pos / KNN;
    int nbr = idx[(size_t)b * NPTS * KNN + pos];
    nb_off[t] = nbr * rs_in;
    cb_off[t] = n * rs_in;
  }

  v8f acc[5];
#pragma unroll
  for (int t = 0; t < 5; ++t) acc[t] = (v8f){0.f, 0.f, 0.f, 0.f, 0.f, 0.f, 0.f, 0.f};

  for (int ks = 0; ks < KSTEPS; ++ks) {
    const int kk = ks * 4;
    v2f a;
#pragma unroll
    for (int v = 0; v < 2; ++v) {
      int c = kk + half2 + v;
      a[v] = (c < K2) ? W[(obase + col) * K2 + c] : 0.f;
    }
#pragma unroll
    for (int t = 0; t < 5; ++t) {
      v2f bb;
#pragma unroll
      for (int v = 0; v < 2; ++v) {
        int c = kk + half2 + v;
        float val;
        if (c < CIN)       val = xb[nb_off[t] + c] - xb[cb_off[t] + c];
        else if (c < K2)   val = xb[cb_off[t] + (c - CIN)];
        else               val = 0.f;
        bb[v] = val;
      }
      acc[t] = wmma_f32(a, bb, acc[t]);
    }
  }

  const int mof = (lane >> 4) << 3;
#pragma unroll
  for (int t = 0; t < 5; ++t)
#pragma unroll
    for (int i = 0; i < 8; ++i)
      lh[(i + mof) * 80 + t * 16 + col] = acc[t][i];
  __syncthreads();

  // 16 o x 8 n outputs; max over k is after the conv, leaky commutes with max.
#pragma unroll
  for (int r = 0; r < 4; ++r) {
    int id = lane + 32 * r;
    int o = id >> 3;
    int nl = id & 7;
    float mx = lh[o * 80 + nl * KNN];
#pragma unroll
    for (int k = 1; k < KNN; ++k) mx = fmaxf(mx, lh[o * 80 + nl * KNN + k]);
    float hv = leakyf(mx);
    int n = nbase + nl;
    if (n < NPTS)
      xc[((size_t)b * NPTS + n) * 256 + choff_out + obase + o] = hv;
  }
}

// ---------------------------------------------------------------------------
// conv5 (256->512) + BN + leaky, fused with max/mean pooling over the 220
// points. One wave per (16-channel tile, batch).
// ---------------------------------------------------------------------------
__global__ __launch_bounds__(32) void conv5_pool_kernel(
    const float* __restrict__ xc, const float* __restrict__ W5,
    const float* __restrict__ g5, const float* __restrict__ b5,
    const float* __restrict__ m5, const float* __restrict__ v5,
    float* __restrict__ p) {
  const int otile = blockIdx.x;  // 0..31
  const int b = blockIdx.y;
  const int lane = threadIdx.x;
  const int col = lane & 15;
  const int half2 = (lane >> 4) << 1;
  const int obase = otile * 16;

  __shared__ float lh[16 * 224];
  __shared__ float redm[32];
  __shared__ float reds[32];

  const float* xb = xc + (size_t)b * NPTS * 256;
  const int mof = (lane >> 4) << 3;
  for (int t = 0; t < 14; ++t) {
    int n = t * 16 + col;
    if (n > NPTS - 1) n = NPTS - 1;
    const float* xrow = xb + (size_t)n * 256;
    const float* wrow = W5 + (size_t)(obase + col) * 256;
    v8f acc = (v8f){0.f, 0.f, 0.f, 0.f, 0.f, 0.f, 0.f, 0.f};
    for (int ks = 0; ks < 64; ++ks) {
      int c = ks * 4 + half2;
      v2f a, bb;
      a[0] = wrow[c];  a[1] = wrow[c + 1];
      bb[0] = xrow[c]; bb[1] = xrow[c + 1];
      acc = wmma_f32(a, bb, acc);
    }
#pragma unroll
    for (int i = 0; i < 8; ++i) lh[(i + mof) * 224 + t * 16 + col] = acc[i];
  }
  __syncthreads();

  const int o = col;
  const int ch = obase + o;
  const float scale = g5[ch] * (1.0f / sqrtf(v5[ch] + EPSB));
  const float mean = m5[ch];
  const float bias = b5[ch];
  const int h = lane >> 4;
  float mx = NEGINF, sm = 0.f;
  for (int n = h * 110; n < h * 110 + 110; ++n) {
    float t = leakyf((lh[o * 224 + n] - mean) * scale + bias);
    mx = fmaxf(mx, t);
    sm += t;
  }
  redm[lane] = mx;
  reds[lane] = sm;
  __syncthreads();
  if (lane < 16) {
    float fmx = fmaxf(redm[lane], redm[lane + 16]);
    float fsm = (reds[lane] + reds[lane + 16]) * (1.0f / 220.0f);
    p[(size_t)b * 1024 + ch] = fmx;
    p[(size_t)b * 1024 + 512 + ch] = fsm;
  }
}

// ---------------------------------------------------------------------------
// FC1: z = leaky(bn(p @ Wl1^T)). One wave per 16x16 output tile, K=1024.
// ---------------------------------------------------------------------------
__global__ __launch_bounds__(32) void fc1_kernel(
    const float* __restrict__ p, const float* __restrict__ Wl1,
    const float* __restrict__ g6, const float* __restrict__ b6,
    const float* __restrict__ m6, const float* __restrict__ v6,
    float* __restrict__ z) {
  const int OT = 64;
  const int btile = blockIdx.x / OT;
  const int otile = blockIdx.x % OT;
  const int lane = threadIdx.x;
  const int col = lane & 15;
  const int half2 = (lane >> 4) << 1;
  int row = btile * 16 + col;
  if (row > NB - 1) row = NB - 1;
  const int o = otile * 16 + col;

  const float* prow = p + (size_t)row * 1024;
  const float* wrow = Wl1 + (size_t)o * 1024;
  v8f acc = (v8f){0.f, 0.f, 0.f, 0.f, 0.f, 0.f, 0.f, 0.f};
  for (int ks = 0; ks < 256; ++ks) {
    int c = ks * 4 + half2;
    v2f a, bb;
    a[0] = prow[c];  a[1] = prow[c + 1];
    bb[0] = wrow[c]; bb[1] = wrow[c + 1];
    acc = wmma_f32(a, bb, acc);
  }
  const int mof = (lane >> 4) << 3;
  const int oc = otile * 16 + col;  // D column index
  const float scale = g6[oc] * (1.0f / sqrtf(v6[oc] + EPSB));
  const float mean = m6[oc];
  const float bias = b6[oc];
#pragma unroll
  for (int i = 0; i < 8; ++i) {
    int r = btile * 16 + i + mof;
    if (r < NB) z[(size_t)r * 1024 + oc] = leakyf((acc[i] - mean) * scale + bias);
  }
}

// ---------------------------------------------------------------------------
// FC2: out = z @ Wl2^T + bl2 (759x380).
// ---------------------------------------------------------------------------
__global__ __launch_bounds__(32) void fc2_kernel(const float* __restrict__ z,
                                                 const float* __restrict__ Wl2,
                                                 const float* __restrict__ bl2,
                                                 float* __restrict__ out) {
  const int OT = 24;
  const int btile = blockIdx.x / OT;
  const int otile = blockIdx.x % OT;
  const int lane = threadIdx.x;
  const int col = lane & 15;
  const int half2 = (lane >> 4) << 1;
  int row = btile * 16 + col;
  if (row > NB - 1) row = NB - 1;
  int o = otile * 16 + col;
  int og = o > 379 ? 379 : o;  // clamp gather, keep EXEC full

  const float* zrow = z + (size_t)row * 1024;
  const float* wrow = Wl2 + (size_t)og * 1024;
  v8f acc = (v8f){0.f, 0.f, 0.f, 0.f, 0.f, 0.f, 0.f, 0.f};
  for (int ks = 0; ks < 256; ++ks) {
    int c = ks * 4 + half2;
    v2f a, bb;
    a[0] = zrow[c];  a[1] = zrow[c + 1];
    bb[0] = wrow[c]; bb[1] = wrow[c + 1];
    acc = wmma_f32(a, bb, acc);
  }
  const int mof = (lane >> 4) << 3;
  const int oc = otile * 16 + col;
  const float bias = (oc < 380) ? bl2[oc] : 0.f;
#pragma unroll
  for (int i = 0; i < 8; ++i) {
    int r = btile * 16 + i + mof;
    if (r < NB && oc < 380) out[(size_t)r * 380 + oc] = acc[i] + bias;
  }
}

// ---------------------------------------------------------------------------
extern "C" void kernel_launch(void* const* d_in, const int* in_sizes, int n_in,
                              void* d_out, int out_size, void* d_ws,
                              size_t ws_size, hipStream_t stream) {
  (void)in_sizes; (void)n_in; (void)out_size; (void)ws_size;
  const float* input = (const float*)d_in[0];
  const float* W1 = (const float*)d_in[1];
  const float* W2 = (const float*)d_in[2];
  const float* W3 = (const float*)d_in[3];
  const float* W4 = (const float*)d_in[4];
  const float* W5 = (const float*)d_in[5];
  const float* g5 = (const float*)d_in[6];
  const float* b5 = (const float*)d_in[7];
  const float* m5 = (const float*)d_in[8];
  const float* v5 = (const float*)d_in[9];
  const float* Wl1 = (const float*)d_in[10];
  const float* g6 = (const float*)d_in[11];
  const float* b6 = (const float*)d_in[12];
  const float* m6 = (const float*)d_in[13];
  const float* v6 = (const float*)d_in[14];
  const float* Wl2 = (const float*)d_in[15];
  const float* bl2 = (const float*)d_in[16];
  float* out = (float*)d_out;

  char* ws = (char*)d_ws;
  float* xc = (float*)ws;  // (759,220,256) channel-concat buffer
  size_t off = (size_t)NB * NPTS * 256 * sizeof(float);
  int* idx = (int*)(ws + off);
  off += (size_t)NB * NPTS * KNN * sizeof(int);
  float* p = (float*)(ws + off);
  off += (size_t)NB * 1024 * sizeof(float);
  float* z = (float*)(ws + off);

  // Stage 1: input (C=3) -> x1 (32 ch) at xc[...,0:32]
  knn_kernel<3><<<NB, 256, 0, stream>>>(input, 3, 0, idx);
  edgeconv_kernel<3, 32><<<dim3(2 * 28, NB), 32, 0, stream>>>(input, 3, 0, W1, idx, xc, 0);
  // Stage 2: x1 -> x2 (32 ch) at xc[...,32:64]
  knn_kernel<32><<<NB, 256, 0, stream>>>(xc, 256, 0, idx);
  edgeconv_kernel<32, 32><<<dim3(2 * 28, NB), 32, 0, stream>>>(xc, 256, 0, W2, idx, xc, 32);
  // Stage 3: x2 -> x3 (64 ch) at xc[...,64:128]
  knn_kernel<32><<<NB, 256, 0, stream>>>(xc, 256, 32, idx);
  edgeconv_kernel<32, 64><<<dim3(4 * 28, NB), 32, 0, stream>>>(xc, 256, 32, W3, idx, xc, 64);
  // Stage 4: x3 -> x4 (128 ch) at xc[...,128:256]
  knn_kernel<64><<<NB, 256, 0, stream>>>(xc, 256, 64, idx);
  edgeconv_kernel<64, 128><<<dim3(8 * 28, NB), 32, 0, stream>>>(xc, 256, 64, W4, idx, xc, 128);
  // conv5 + bn + leaky + max/mean pool -> p (759,1024)
  conv5_pool_kernel<<<dim3(32, NB), 32, 0, stream>>>(xc, W5, g5, b5, m5, v5, p);
  // fc1 + bn + leaky -> z (759,1024)
  fc1_kernel<<<48 * 64, 32, 0, stream>>>(p, Wl1, g6, b6, m6, v6, z);
  // fc2 + bias -> out (759,380)
  fc2_kernel<<<48 * 24, 32, 0, stream>>>(z, Wl2, bl2, out);
}